// ComplexBatchNormal_1348619731485
// MI455X (gfx1250) — compile-verified
//
#include <hip/hip_runtime.h>
#include <stdint.h>

// ComplexBatchNorm for x:(16,32,256,256) fp32, params:(32,256,256) fp32,
// output complex64 interleaved (re,im) pairs.
//
// Memory-bound: ~570MB min traffic -> ~24.5us floor at 23.3 TB/s.
// Single pass: batch (B=16) held in registers per thread, exact centered
// statistics, NT loads/stores, TDM (tensor_load_to_lds) staging of the
// per-position affine params into LDS.

typedef float        f32x2 __attribute__((ext_vector_type(2)));
typedef float        f32x4 __attribute__((ext_vector_type(4)));
typedef unsigned int u32x4 __attribute__((ext_vector_type(4)));
typedef int          i32x4 __attribute__((ext_vector_type(4)));
typedef int          i32x8 __attribute__((ext_vector_type(8)));

#define EPSV 1e-5f

static constexpr int BATCH = 16;
static constexpr int CHW   = 32 * 256 * 256;   // 2,097,152 positions
static constexpr int TPB   = 256;              // 8 waves (wave32)
static constexpr int VEC   = 2;                // chw positions per thread
static constexpr int EPB   = TPB * VEC;        // 512 positions per block

__global__ __launch_bounds__(TPB) void complex_bn_kernel(
    const float* __restrict__ xr,  const float* __restrict__ xi,
    const float* __restrict__ grr, const float* __restrict__ gri,
    const float* __restrict__ gii, const float* __restrict__ gbeta,
    float* __restrict__ out)
{
    __shared__ float lds_params[4 * EPB];      // 8 KB: grr | gri | gii | beta

    const int base = (int)blockIdx.x * EPB;

    // ---- Stage the four parameter tiles into LDS via the Tensor Data Mover.
#if __has_builtin(__builtin_amdgcn_tensor_load_to_lds)
    if (threadIdx.x < 32) {                    // wave 0 issues the DMAs
        const float* srcs[4] = { grr, gri, gii, gbeta };
#pragma unroll
        for (int p = 0; p < 4; ++p) {
            const uint64_t ga = (uint64_t)(uintptr_t)(srcs[p] + base);
            // D# group 0: count=1, lds_addr, global_addr[56:0], type=2
            u32x4 g0;
            g0.x = 1u;
            g0.y = (unsigned)(p * EPB * 4);                      // LDS byte addr
            g0.z = (unsigned)(ga & 0xffffffffu);                 // gaddr[31:0]
            g0.w = (unsigned)((ga >> 32) & 0x01ffffffu) | 0x80000000u; // [56:32]|type=2
            // D# group 1: data_size=4B, tensor_dim0=CHW, tensor_dim1=1,
            //             tile_dim0=EPB (1-D tile), dim0_stride=CHW
            i32x8 g1;
            g1[0] = (int)(2u << 16);                             // data_size=4B
            g1[1] = (int)(((unsigned)CHW & 0xffffu) << 16);      // dim0[15:0]
            g1[2] = (int)((((unsigned)CHW >> 16) & 0xffffu) | (1u << 16)); // dim0 hi | dim1 lo
            g1[3] = (int)((unsigned)EPB << 16);                  // tile_dim0
            g1[4] = 0;                                           // tile_dim1/2 unused
            g1[5] = (int)CHW;                                    // dim0_stride lo
            g1[6] = 0;
            g1[7] = 0;
            i32x4 z4 = {0, 0, 0, 0};
#if __has_include(<hip/amd_detail/amd_gfx1250_TDM.h>)
            i32x8 z8 = {0, 0, 0, 0, 0, 0, 0, 0};
            __builtin_amdgcn_tensor_load_to_lds(g0, g1, z4, z4, z8, 0);
#else
            __builtin_amdgcn_tensor_load_to_lds(g0, g1, z4, z4, 0);
#endif
        }
        __builtin_amdgcn_s_wait_tensorcnt(0);
    }
    __syncthreads();
#else
    for (int t = threadIdx.x; t < EPB; t += TPB) {
        lds_params[0 * EPB + t] = grr[base + t];
        lds_params[1 * EPB + t] = gri[base + t];
        lds_params[2 * EPB + t] = gii[base + t];
        lds_params[3 * EPB + t] = gbeta[base + t];
    }
    __syncthreads();
#endif

    // The TDM writes LDS through a descriptor (numeric offset, not a pointer),
    // so alias analysis thinks lds_params is never written and folds the reads
    // to poison, killing the whole compute chain. Escape the pointer with a
    // memory clobber so the compiler treats the staged data as live.
    {
        float* p = &lds_params[0];
        asm volatile("" : "+r"(p) : : "memory");
    }

    const int idx = base + (int)threadIdx.x * VEC;

    // ---- Load the whole batch for our 2 positions (32 NT b64 loads in flight).
    f32x2 r[BATCH], m[BATCH];
#pragma unroll
    for (int b = 0; b < BATCH; ++b) {
        const size_t off = (size_t)b * CHW + (size_t)idx;
        r[b] = __builtin_nontemporal_load((const f32x2*)(xr + off));
        m[b] = __builtin_nontemporal_load((const f32x2*)(xi + off));
    }

    // ---- Exact batch statistics (matches reference two-pass formulation).
    f32x2 sr = {0.f, 0.f}, si = {0.f, 0.f};
#pragma unroll
    for (int b = 0; b < BATCH; ++b) { sr += r[b]; si += m[b]; }
    const f32x2 mur = sr * 0.0625f;
    const f32x2 mui = si * 0.0625f;

    f32x2 vrr = {0.f, 0.f}, vii = {0.f, 0.f}, vri = {0.f, 0.f};
#pragma unroll
    for (int b = 0; b < BATCH; ++b) {
        const f32x2 a = r[b] - mur;
        const f32x2 c = m[b] - mui;
        vrr += a * a; vii += c * c; vri += a * c;
    }
    vrr = vrr * 0.0625f + EPSV;
    vii = vii * 0.0625f + EPSV;
    vri = vri * 0.0625f;

    // ---- Analytic inverse sqrt of the 2x2 covariance.
    f32x2 Wrr, Wii, Wri;
#pragma unroll
    for (int k = 0; k < 2; ++k) {
        const float s  = __builtin_sqrtf(vrr[k] * vii[k] - vri[k] * vri[k]);
        const float t  = __builtin_sqrtf(vrr[k] + vii[k] + 2.0f * s);
        const float iv = 1.0f / (s * t);
        Wrr[k] = (vii[k] + s) * iv;
        Wii[k] = (vrr[k] + s) * iv;
        Wri[k] = -vri[k] * iv;
    }

    // ---- Affine params from LDS (TDM-staged).
    const int l = (int)threadIdx.x * VEC;
    const f32x2 Grr = *(const f32x2*)&lds_params[0 * EPB + l];
    const f32x2 Gri = *(const f32x2*)&lds_params[1 * EPB + l];
    const f32x2 Gii = *(const f32x2*)&lds_params[2 * EPB + l];
    const f32x2 Bt  = *(const f32x2*)&lds_params[3 * EPB + l];

    // ---- Normalize + affine, write interleaved complex64 (one b128/batch).
#pragma unroll
    for (int b = 0; b < BATCH; ++b) {
        const f32x2 rc = r[b] - mur;
        const f32x2 ic = m[b] - mui;
        const f32x2 nr = Wrr * rc + Wri * ic;
        const f32x2 ni = Wii * ic + Wri * rc;
        const f32x2 br = Grr * nr + Gri * ni + Bt;
        const f32x2 bi = Gri * nr + Gii * ni + Bt;
        const f32x4 o = { br.x, bi.x, br.y, bi.y };
        float* op = out + 2u * ((size_t)b * CHW + (size_t)idx);
        __builtin_nontemporal_store(o, (f32x4*)op);
    }
}

extern "C" void kernel_launch(void* const* d_in, const int* in_sizes, int n_in,
                              void* d_out, int out_size, void* d_ws, size_t ws_size,
                              hipStream_t stream) {
    (void)in_sizes; (void)n_in; (void)out_size; (void)d_ws; (void)ws_size;
    const float* xr  = (const float*)d_in[0];
    const float* xi  = (const float*)d_in[1];
    const float* grr = (const float*)d_in[2];
    const float* gri = (const float*)d_in[3];
    const float* gii = (const float*)d_in[4];
    const float* gb  = (const float*)d_in[5];
    float* out = (float*)d_out;

    const int blocks = CHW / EPB;   // 4096 blocks of 256 threads (8 waves)
    complex_bn_kernel<<<blocks, TPB, 0, stream>>>(xr, xi, grr, gri, gii, gb, out);
}